// RGCN_4587025072288
// MI455X (gfx1250) — compile-verified
//
#include <hip/hip_runtime.h>

// ---------------------------------------------------------------------------
// RGCN (2-layer, mean aggregation) for MI455X / gfx1250, fp32 end-to-end.
//
// Key optimizations (reasoned from MI455X specs, compile-only loop):
//  * Layer-2 linearity: mean_e(x[src]@W2[r]) = (mean_e x[src]) @ W2[r]
//    -> never materialize xW[R,N,H]; dense batched WMMA GEMM at the end.
//  * Scatter split into 2 relation groups so the atomic destination
//    (102.4 MB/group) stays resident in the 192 MB L2 (atomics RMW in L2).
//  * Non-temporal loads for the streamed W1/hbar gathers so they don't
//    evict the L2-resident accumulators.
//  * Mean-normalize fused into GEMM A-tile staging (saves 410 MB HBM).
// ---------------------------------------------------------------------------

typedef float v2f __attribute__((ext_vector_type(2)));
typedef float v4f __attribute__((ext_vector_type(4)));
typedef float v8f __attribute__((ext_vector_type(8)));

#define HID 128
#define LDSA_STRIDE 132   // 33 float4s: bank-conflict-free rows, 16B aligned

// ---------------- utility: zero a float buffer (grid-stride) ---------------
__global__ void zero_f32(float* __restrict__ p, long long n) {
  long long i = (long long)blockIdx.x * blockDim.x + threadIdx.x;
  long long stride = (long long)gridDim.x * blockDim.x;
  for (; i < n; i += stride) p[i] = 0.0f;
}

// ---- scatter rows into sums[(rel*N+dst), :] with atomics, float4/thread ---
// 32 threads per edge; only edges with rel in [relLo, relHi) are processed
// (two passes keep each pass's atomic working set inside L2).
// rowsByRel=1: rows indexed by (rel*N+src), streamed NT (layer 1, W1).
// rowsByRel=0: rows indexed by src, regular loads (layer 2, x is L2-hot).
__global__ void scatter_rows(const float* __restrict__ rows,
                             const int* __restrict__ src,
                             const int* __restrict__ dst,
                             const int* __restrict__ etype,
                             float* __restrict__ sums,
                             float* __restrict__ cnt,
                             int E, int N, int rowsByRel,
                             int relLo, int relHi) {
  long long idx = (long long)blockIdx.x * blockDim.x + threadIdx.x;
  int e = (int)(idx >> 5);
  int q = (int)(idx & 31);          // quad id: columns q*4 .. q*4+3
  if (e >= E) return;
  int r = etype[e];
  if (r < relLo || r >= relHi) return;
  int s = src[e];
  int d = dst[e];
  long long srcRow = rowsByRel ? ((long long)r * N + s) : (long long)s;
  const v4f* sp = (const v4f*)&rows[srcRow * HID + q * 4];
  v4f v = rowsByRel ? __builtin_nontemporal_load(sp) : *sp;
  float* dp = &sums[((long long)r * N + d) * HID + q * 4];
  atomicAdd(dp + 0, v.x);
  atomicAdd(dp + 1, v.y);
  atomicAdd(dp + 2, v.z);
  atomicAdd(dp + 3, v.w);
  if (cnt != nullptr && q == 0) atomicAdd(&cnt[(long long)r * N + d], 1.0f);
}

// ---------------- invc[rn] = 1 / max(cnt[rn], 1) ---------------------------
__global__ void inv_cnt(const float* __restrict__ cnt,
                        float* __restrict__ invc, long long n) {
  long long idx = (long long)blockIdx.x * blockDim.x + threadIdx.x;
  if (idx >= n) return;
  invc[idx] = 1.0f / fmaxf(cnt[idx], 1.0f);
}

// --------- x = leaky_relu( sum_r sums[r]*invc[r] + root1 + b1 ) ------------
__global__ void compute_x(const float* __restrict__ sums,
                          const float* __restrict__ invc,
                          const float* __restrict__ root1,
                          const float* __restrict__ b1,
                          float* __restrict__ x, int N, int R) {
  long long idx = (long long)blockIdx.x * blockDim.x + threadIdx.x;
  if (idx >= (long long)N * HID) return;
  int n = (int)(idx >> 7);
  int h = (int)(idx & (HID - 1));
  float acc = 0.0f;
  for (int r = 0; r < R; ++r) {
    long long rn = (long long)r * N + n;
    acc += __builtin_nontemporal_load(&sums[rn * HID + h]) * invc[rn];
  }
  float v = acc + root1[idx] + b1[h];
  x[idx] = (v >= 0.0f) ? v : 0.01f * v;   // leaky_relu, NEG_SLOPE=0.01
}

// ---------------------------------------------------------------------------
// out[n, :] = sum_r (sums[r,n,:]*invc[r,n]) @ W2[r]  +  x[n,:] @ root2  +  b2
// 1 block = 256 threads = 8 wave32 = one 16x128 output tile (3125 blocks).
// Each wave: one 16x16 tile, K=128 via 32x V_WMMA_F32_16X16X4_F32 per matrix,
// 9 matrices (8 relations + root2). Mean-normalization fused into A staging.
// A staged in LDS via NT float4 loads; B streamed from global (576 KB total,
// L2-hot). No divergence: EXEC stays all-ones for WMMA.
// ---------------------------------------------------------------------------
__global__ __launch_bounds__(256)
void rgcn_out_gemm(const float* __restrict__ sums,   // [R*N, H] raw segment sums
                   const float* __restrict__ invc,   // [R*N] 1/max(cnt,1)
                   const float* __restrict__ x,      // [N, H]
                   const float* __restrict__ W2,     // [R, H, H]
                   const float* __restrict__ root2,  // [H, H]
                   const float* __restrict__ b2,     // [H]
                   float* __restrict__ out,          // [N, H]
                   int N, int R) {
  __shared__ float lA[16 * LDSA_STRIDE];

  const int tid = threadIdx.x;
  const int lane = tid & 31;
  const int wave = tid >> 5;            // 0..7
  const int rowBase = blockIdx.x * 16;
  const int colBase = wave * 16;

  const int m15 = lane & 15;            // M (for A/C) or N-col (for B/C)
  const int khalf = (lane < 16) ? 0 : 2;

  v8f c = {};

  for (int m = 0; m <= R; ++m) {        // m<R: relation m; m==R: root2 term
    const float* Asrc = (m < R) ? (sums + ((long long)m * N + rowBase) * HID)
                                : (x + (long long)rowBase * HID);
    // Cooperative stage of the 16x128 A tile into LDS, scaled by invc
    // (fused mean). 512 float4s; 2 per thread. NT: streamed once.
    #pragma unroll
    for (int j = 0; j < 2; ++j) {
      int li = tid + j * 256;           // 0..511
      int rr = li >> 5;                 // row in tile 0..15
      int c4 = (li & 31) * 4;           // col 0,4,...,124
      int gr = rowBase + rr;
      int crr = (gr < N) ? rr : 0;      // clamp tail rows
      v4f v = __builtin_nontemporal_load(
          (const v4f*)&Asrc[(long long)crr * HID + c4]);
      float scl = (m < R) ? invc[(long long)m * N + rowBase + crr] : 1.0f;
      v *= scl;
      *(v4f*)&lA[rr * LDSA_STRIDE + c4] = v;
    }
    __syncthreads();

    const float* B = (m < R) ? (W2 + (long long)m * HID * HID) : root2;
    #pragma unroll 4
    for (int k0 = 0; k0 < HID; k0 += 4) {
      int kb = k0 + khalf;
      // A frag: lane m15 = M row; VGPR pair = K {kb, kb+1}
      v2f a = *(const v2f*)&lA[m15 * LDSA_STRIDE + kb];
      // B frag: rows kb / kb+1 striped across lanes (N col = colBase+m15)
      v2f b;
      int col = colBase + m15;
      b.x = B[(long long)kb * HID + col];
      b.y = B[(long long)(kb + 1) * HID + col];
      c = __builtin_amdgcn_wmma_f32_16x16x4_f32(
          /*neg_a=*/false, a, /*neg_b=*/false, b,
          /*c_mod=*/(short)0, c, /*reuse_a=*/false, /*reuse_b=*/false);
    }
    __syncthreads();
  }

  // Epilogue: C/D layout = 8 VGPRs, lanes 0-15 -> M=i, lanes 16-31 -> M=i+8.
  int col = colBase + m15;
  float bias = b2[col];
  int rofs = (lane < 16) ? 0 : 8;
  #pragma unroll
  for (int i = 0; i < 8; ++i) {
    int row = rowBase + rofs + i;
    if (row < N)
      __builtin_nontemporal_store(c[i] + bias, &out[(long long)row * HID + col]);
  }
}

// ---------------------------------------------------------------------------
extern "C" void kernel_launch(void* const* d_in, const int* in_sizes, int n_in,
                              void* d_out, int out_size, void* d_ws, size_t ws_size,
                              hipStream_t stream) {
  const int* edge_index = (const int*)d_in[0];   // [2, E]
  const int* edge_type  = (const int*)d_in[1];   // [E]
  const float* W1    = (const float*)d_in[2];    // [R, N, H]
  const float* root1 = (const float*)d_in[3];    // [N, H]
  const float* b1    = (const float*)d_in[4];    // [H]
  const float* W2    = (const float*)d_in[5];    // [R, H, H]
  const float* root2 = (const float*)d_in[6];    // [H, H]
  const float* b2    = (const float*)d_in[7];    // [H]
  float* out = (float*)d_out;

  const int E = in_sizes[1];
  const int H = in_sizes[4];                      // 128 (kernels hardcode HID)
  const int N = in_sizes[3] / H;                  // 50000
  const int R = in_sizes[5] / (H * H);            // 8
  (void)n_in; (void)ws_size; (void)out_size; (void)H;

  const int* src = edge_index;                    // row 0
  const int* dst = edge_index + E;                // row 1

  // Workspace (floats): sums[R*N*H] | cnt[R*N] | invc[R*N] | x[N*H] (~234 MB)
  float* sums = (float*)d_ws;
  long long sumsN = (long long)R * N * HID;
  float* cnt  = sums + sumsN;
  float* invc = cnt + (long long)R * N;
  float* x    = invc + (long long)R * N;

  const int T = 256;
  long long edgeWork = (long long)E * 32;         // 32 threads (float4s) / edge
  int edgeBlocks = (int)((edgeWork + T - 1) / T);
  const int Rmid = (R + 1) / 2;                   // relation-group split point

  // 1. zero sums + counts (contiguous)
  zero_f32<<<4096, T, 0, stream>>>(sums, sumsN + (long long)R * N);
  // 2. layer-1 scatter in 2 relation groups (atomic set fits 192 MB L2)
  scatter_rows<<<edgeBlocks, T, 0, stream>>>(W1, src, dst, edge_type,
                                             sums, cnt, E, N, 1, 0, Rmid);
  scatter_rows<<<edgeBlocks, T, 0, stream>>>(W1, src, dst, edge_type,
                                             sums, cnt, E, N, 1, Rmid, R);
  // 3. reciprocal counts (shared by both layers; segments are identical)
  inv_cnt<<<(int)(((long long)R * N + T - 1) / T), T, 0, stream>>>(
      cnt, invc, (long long)R * N);
  // 4. x = leaky_relu( sum_r mean_r + root1 + b1 )
  compute_x<<<(int)(((long long)N * HID + T - 1) / T), T, 0, stream>>>(
      sums, invc, root1, b1, x, N, R);
  // 5. re-zero sums
  zero_f32<<<4096, T, 0, stream>>>(sums, sumsN);
  // 6. layer-2 scatter (x rows; x is L2-hot), 2 relation groups
  scatter_rows<<<edgeBlocks, T, 0, stream>>>(x, src, dst, edge_type,
                                             sums, nullptr, E, N, 0, 0, Rmid);
  scatter_rows<<<edgeBlocks, T, 0, stream>>>(x, src, dst, edge_type,
                                             sums, nullptr, E, N, 0, Rmid, R);
  // 7. dense WMMA GEMM with fused mean: out = sum_r (sums*invc)@W2 + x@root2 + b2
  rgcn_out_gemm<<<(N + 15) / 16, T, 0, stream>>>(sums, invc, x, W2, root2, b2,
                                                 out, N, R);
}